// HGNNClassifier_37228776522459
// MI455X (gfx1250) — compile-verified
//
#include <hip/hip_runtime.h>

// ---------------------------------------------------------------------------
// HGNN classifier: out = HGConv(relu(HGConv(x, W1, b1)), W2, b2)
// HGConv(x) = Dinv * scatter_n( Binv * scatter_e( x @ W ) ) + b
// (Binv/Dinv are constant per segment -> applied after segment-sum, exact.)
// Dense GEMMs use V_WMMA_F32_16X16X4_F32 (full-f32 WMMA path on gfx1250).
// Scatter phases are HBM/atomic bound; GEMM precision kept at f32 since the
// matrix work is ~3.7 GFLOP vs ~4 GB of message traffic.
// ---------------------------------------------------------------------------

typedef __attribute__((ext_vector_type(2))) float v2f;
typedef __attribute__((ext_vector_type(8))) float v8f;

#define HG_N_NODES   100000
#define HG_N_EDGES   50000
#define HG_NNZ       1600000
#define HG_IN_DIM    128
#define HG_HID_DIM   128
#define HG_NCLS      16

// ---------------------------------------------------------------- utilities
__global__ void hg_zero_f32(float* __restrict__ p, long n) {
    long i = (long)blockIdx.x * blockDim.x + threadIdx.x;
    long stride = (long)gridDim.x * blockDim.x;
    for (; i < n; i += stride) p[i] = 0.0f;
}

__global__ void hg_degree(const int* __restrict__ node_idx,
                          const int* __restrict__ edge_idx,
                          float* __restrict__ deg, float* __restrict__ edeg,
                          int nnz) {
    int i = blockIdx.x * blockDim.x + threadIdx.x;
    if (i < nnz) {
        atomicAdd(&deg[node_idx[i]], 1.0f);
        atomicAdd(&edeg[edge_idx[i]], 1.0f);
    }
}

__global__ void hg_recip(float* __restrict__ p, int n) {
    int i = blockIdx.x * blockDim.x + threadIdx.x;
    if (i < n) {
        float v = p[i];
        p[i] = (v > 0.0f) ? (1.0f / v) : 0.0f;
    }
}

// ------------------------------------------------------------- WMMA f32 GEMM
// out[M x NDIM] = A[M x 128] * W[128 x NDIM], row-major, K fixed at 128.
// One wave computes one 16x16 tile via 32x V_WMMA_F32_16X16X4_F32.
template <int NDIM>
__global__ void hg_gemm_wmma(const float* __restrict__ A,
                             const float* __restrict__ W,
                             float* __restrict__ out, int M) {
    const int lane = threadIdx.x & 31;
    const int wave = threadIdx.x >> 5;
    const int lo   = lane & 15;   // N index within tile / M index for A
    const int hi   = lane >> 4;   // K-half select (0: K0/K1, 1: K2/K3)

    constexpr int NT = NDIM / 16;                       // N tiles
    const int gwave = blockIdx.x * (blockDim.x >> 5) + wave;
    const int mtile = gwave / NT;
    const int ntile = gwave % NT;
    if (mtile * 16 >= M) return;                        // wave-uniform guard

    const int mbase = mtile * 16;
    const int nbase = ntile * 16;

    const float* __restrict__ arow = A + (long)(mbase + lo) * 128;

    v8f c = {};
#pragma unroll 8
    for (int k0 = 0; k0 < 128; k0 += 4) {
        const int ka = k0 + 2 * hi;
        v2f a, b;
        // A 16x4 f32 layout: lanes 0-15 hold M=lo,K={k0,k0+1}; lanes 16-31 K={k0+2,k0+3}
        a.x = arow[ka];
        a.y = arow[ka + 1];
        // B 4x16 f32: rows striped across lanes within a VGPR, mirrored K split
        b.x = W[(long)ka * NDIM + nbase + lo];
        b.y = W[(long)(ka + 1) * NDIM + nbase + lo];
        c = __builtin_amdgcn_wmma_f32_16x16x4_f32(
                /*neg_a=*/false, a, /*neg_b=*/false, b,
                /*c_mod=*/(short)0, c, /*reuse_a=*/false, /*reuse_b=*/false);
    }

    // C/D layout: VGPR v -> M = v + 8*hi ; lane lo -> N
#pragma unroll
    for (int v = 0; v < 8; ++v)
        out[(long)(mbase + v + 8 * hi) * NDIM + nbase + lo] = c[v];
}

// ----------------------------------------------------- segment scatter (add)
// dst[dst_idx[i]][:] += src[src_idx[i]][:]   (F threads cooperate per nnz)
template <int F>
__global__ void hg_scatter_add(const float* __restrict__ src,
                               float* __restrict__ dst,
                               const int* __restrict__ src_idx,
                               const int* __restrict__ dst_idx, int nnz) {
    constexpr int PER_BLOCK = 256 / F;
    const int t = threadIdx.x;
    const int g = t / F;
    const int f = t % F;
    const int i = blockIdx.x * PER_BLOCK + g;
    if (i >= nnz) return;
    const int s = src_idx[i];
    const int d = dst_idx[i];
    atomicAdd(&dst[(long)d * F + f], src[(long)s * F + f]);
}

template <int F>
__global__ void hg_scale_rows(float* __restrict__ p,
                              const float* __restrict__ scale, int rows) {
    long i = (long)blockIdx.x * blockDim.x + threadIdx.x;
    if (i < (long)rows * F) p[i] *= scale[i / F];
}

template <int F, bool RELU>
__global__ void hg_finalize_rows(float* __restrict__ p,
                                 const float* __restrict__ scale,
                                 const float* __restrict__ bias, int rows) {
    long i = (long)blockIdx.x * blockDim.x + threadIdx.x;
    if (i < (long)rows * F) {
        float v = scale[i / F] * p[i] + bias[i % F];
        if (RELU) v = (v > 0.0f) ? v : 0.0f;
        p[i] = v;
    }
}

// ------------------------------------------------------------------- driver
extern "C" void kernel_launch(void* const* d_in, const int* in_sizes, int n_in,
                              void* d_out, int out_size, void* d_ws, size_t ws_size,
                              hipStream_t stream) {
    const float* x        = (const float*)d_in[0];
    const int*   node_idx = (const int*)  d_in[1];
    const int*   edge_idx = (const int*)  d_in[2];
    // d_in[3] = num_edges scalar (compile-time constant HG_N_EDGES)
    const float* W1       = (const float*)d_in[4];
    const float* b1       = (const float*)d_in[5];
    const float* W2       = (const float*)d_in[6];
    const float* b2       = (const float*)d_in[7];
    float* out = (float*)d_out;

    // workspace layout (floats)
    float* ws    = (float*)d_ws;
    float* deg   = ws;                                   // [100000]  -> Dinv
    float* edeg  = ws + HG_N_NODES;                      // [50000]   -> Binv
    float* xw    = edeg + HG_N_EDGES;                    // [100000*128] (reused as h)
    float* efeat = xw + (long)HG_N_NODES * HG_HID_DIM;   // [50000*128]
    float* hw    = efeat + (long)HG_N_EDGES * HG_HID_DIM;// [100000*16]
    float* e2    = hw + (long)HG_N_NODES * HG_NCLS;      // [50000*16]

    const int ZB = 256;
    auto zgrid = [](long n) { long g = (n + 255) / 256; return (int)(g > 8192 ? 8192 : g); };

    // 1) zero accumulators (deg+edeg contiguous; efeat..e2 contiguous; out)
    hg_zero_f32<<<zgrid(HG_N_NODES + HG_N_EDGES), ZB, 0, stream>>>(deg, HG_N_NODES + HG_N_EDGES);
    long acc_n = (long)HG_N_EDGES * HG_HID_DIM + (long)HG_N_NODES * HG_NCLS
               + (long)HG_N_EDGES * HG_NCLS;
    hg_zero_f32<<<zgrid(acc_n), ZB, 0, stream>>>(efeat, acc_n);
    hg_zero_f32<<<zgrid((long)HG_N_NODES * HG_NCLS), ZB, 0, stream>>>(out, (long)HG_N_NODES * HG_NCLS);

    // 2) degrees, 3) reciprocals (deg & edeg are contiguous -> one recip pass)
    hg_degree<<<(HG_NNZ + 255) / 256, 256, 0, stream>>>(node_idx, edge_idx, deg, edeg, HG_NNZ);
    hg_recip<<<(HG_N_NODES + HG_N_EDGES + 255) / 256, 256, 0, stream>>>(deg, HG_N_NODES + HG_N_EDGES);

    // ---- layer 1 ----
    // 4) xw = x @ W1   (6250 M-tiles x 8 N-tiles, 8 waves/block)
    {
        int mtiles = (HG_N_NODES + 15) / 16;            // 6250
        int waves  = mtiles * (HG_HID_DIM / 16);        // 50000
        hg_gemm_wmma<HG_HID_DIM><<<(waves + 7) / 8, 256, 0, stream>>>(x, W1, xw, HG_N_NODES);
    }
    // 5) node -> edge scatter (raw segment sum)
    hg_scatter_add<128><<<(HG_NNZ + 1) / 2, 256, 0, stream>>>(xw, efeat, node_idx, edge_idx, HG_NNZ);
    // 6) e_feat *= Binv
    hg_scale_rows<128><<<((long)HG_N_EDGES * 128 + 255) / 256, 256, 0, stream>>>(efeat, edeg, HG_N_EDGES);
    // 7) reuse xw as node accumulator
    hg_zero_f32<<<zgrid((long)HG_N_NODES * 128), ZB, 0, stream>>>(xw, (long)HG_N_NODES * 128);
    // 8) edge -> node scatter
    hg_scatter_add<128><<<(HG_NNZ + 1) / 2, 256, 0, stream>>>(efeat, xw, edge_idx, node_idx, HG_NNZ);
    // 9) h = relu(Dinv * raw + b1)
    hg_finalize_rows<128, true><<<((long)HG_N_NODES * 128 + 255) / 256, 256, 0, stream>>>(xw, deg, b1, HG_N_NODES);

    // ---- layer 2 ----
    // 10) hw = h @ W2
    {
        int mtiles = (HG_N_NODES + 15) / 16;            // 6250 (1 N-tile)
        hg_gemm_wmma<HG_NCLS><<<(mtiles + 7) / 8, 256, 0, stream>>>(xw, W2, hw, HG_N_NODES);
    }
    // 11) node -> edge scatter (16-dim)
    hg_scatter_add<16><<<(HG_NNZ + 15) / 16, 256, 0, stream>>>(hw, e2, node_idx, edge_idx, HG_NNZ);
    // 12) e2 *= Binv
    hg_scale_rows<16><<<((long)HG_N_EDGES * 16 + 255) / 256, 256, 0, stream>>>(e2, edeg, HG_N_EDGES);
    // 13) edge -> node scatter into d_out
    hg_scatter_add<16><<<(HG_NNZ + 15) / 16, 256, 0, stream>>>(e2, out, edge_idx, node_idx, HG_NNZ);
    // 14) out = Dinv * raw + b2
    hg_finalize_rows<16, false><<<((long)HG_N_NODES * 16 + 255) / 256, 256, 0, stream>>>(out, deg, b2, HG_N_NODES);
}